// ConvNetIzhikevich_33225867001929
// MI455X (gfx1250) — compile-verified
//
#include <hip/hip_runtime.h>
#include <hip/hip_bf16.h>

// Fused 3x3 SAME conv + Izhikevich scan, one layer per launch.
// Tensor: [T=32, B=16, 1, 256, 256]. Inter-layer spikes stored as u8 {0,1}.
// Each wave32 owns one 16x16 tile of one image; (v,u) state lives in VGPRs
// across the t-loop (WMMA C-layout: VGPR j <-> row j/j+8, lane&15 <-> col).
// Layer 0 (fp32 input): double-buffered GLOBAL_LOAD_ASYNC_TO_LDS staging.

typedef __attribute__((ext_vector_type(2))) float v2f;
typedef __attribute__((ext_vector_type(8))) float v8f;

#define T_STEPS 32
#define BATCH   16
#define IMG     256
#define LDS_W   20      // 18 valid cols + 2 zero pad (K rounded to 20)
#define LDS_H   18
#define SLICE   (LDS_H * LDS_W)          // 360
#define NCHUNK  12                       // ceil(360/32) async/stage chunks per lane
#define WSLICE  (2 * SLICE + 32)         // 2 buffers + 32-float dump area

#if defined(__gfx1250__) && __has_builtin(__builtin_amdgcn_wmma_f32_16x16x4_f32)
#define USE_WMMA 1
#else
#define USE_WMMA 0
#endif

#if __has_builtin(__builtin_amdgcn_global_load_async_to_lds_b32)
#define HAS_ASYNC 1
#else
#define HAS_ASYNC 0
#endif

__device__ __forceinline__ float bandw(int d, float w0, float w1, float w2) {
    // B[k][n] = w[k-n] for k-n in {0,1,2}, else 0  (3-tap band matrix)
    return d == 0 ? w0 : (d == 1 ? w1 : (d == 2 ? w2 : 0.0f));
}

template <typename InT, typename OutT>
__global__ __launch_bounds__(256) void izhi_layer(const InT* __restrict__ in,
                                                  OutT* __restrict__ out,
                                                  const float* __restrict__ w) {
    __shared__ float sl[8][WSLICE];      // per-wave: 2x(18x20) f32 + dump (23.5 KB)

    const int lane = threadIdx.x & 31;
    const int wid  = threadIdx.x >> 5;
    const int tile = blockIdx.x * 8 + wid;          // 512 blocks * 8 waves = 4096 tiles
    const int b    = tile >> 8;                      // 256 tiles per image
    const int rem  = tile & 255;
    const int tileY = rem >> 4, tileX = rem & 15;
    const int rowBase = tileY * 16 - 1;              // LDS row 0 = global row -1 (halo)
    const int colBase = tileX * 16 - 1;
    const int half = (lane >> 4) & 1;                // lane-half selects K/M split
    const int n    = lane & 15;                      // output column within tile

    float W9[9];
#pragma unroll
    for (int i = 0; i < 9; ++i) W9[i] = w[i];

#if USE_WMMA
    // Banded B operands: bw[dr][kc] = { B[k0][n], B[k0+1][n] },
    // k0 = kc*4 + half*2 (f32 16x16x4 B layout: K split across lane halves).
    v2f bw[3][5];
#pragma unroll
    for (int dr = 0; dr < 3; ++dr) {
        float w0 = W9[dr * 3 + 0], w1 = W9[dr * 3 + 1], w2 = W9[dr * 3 + 2];
#pragma unroll
        for (int kc = 0; kc < 5; ++kc) {
            int k0 = kc * 4 + half * 2;
            bw[dr][kc].x = bandw(k0 - n,     w0, w1, w2);
            bw[dr][kc].y = bandw(k0 + 1 - n, w0, w1, w2);
        }
    }
#endif

    // Izhikevich state for this lane's 8 pixels (rows j + half*8, col n)
    float v[8], u[8];
#pragma unroll
    for (int j = 0; j < 8; ++j) { v[j] = -70.0f; u[j] = -14.0f; }

    float* s0   = sl[wid];               // staging buffer 0
    float* s1   = s0 + SLICE;            // staging buffer 1
    float* dump = s0 + 2 * SLICE;        // 32-float scribble area for OOB lanes

    // conv (WMMA banded matmul) + neuron update + spike store for one timestep
    auto step = [&](const float* s, size_t base) {
        v8f c = {0.f, 0.f, 0.f, 0.f, 0.f, 0.f, 0.f, 0.f};
#if USE_WMMA
#pragma unroll
        for (int dr = 0; dr < 3; ++dr) {
#pragma unroll
            for (int kc = 0; kc < 5; ++kc) {
                int k0 = kc * 4 + half * 2;
                v2f a = *(const v2f*)&s[((lane & 15) + dr) * LDS_W + k0];
                c = __builtin_amdgcn_wmma_f32_16x16x4_f32(
                        false, a, false, bw[dr][kc], (short)0, c, false, false);
            }
        }
#else
#pragma unroll
        for (int j = 0; j < 8; ++j) {
            int mrow = j + half * 8;
            float acc = 0.0f;
#pragma unroll
            for (int dr = 0; dr < 3; ++dr)
#pragma unroll
                for (int dc = 0; dc < 3; ++dc)
                    acc += s[(mrow + dr) * LDS_W + (n + dc)] * W9[dr * 3 + dc];
            c[j] = acc;
        }
#endif
        // Izhikevich update (k = TAU_INV*DT = 0.25) + spike/reset
#pragma unroll
        for (int j = 0; j < 8; ++j) {
            float cur = c[j];
            float vv = v[j], uu = u[j];
            float vn = vv + 0.25f * (0.04f * vv * vv + 5.0f * vv + 140.0f - uu + cur);
            float un = uu + 0.005f * (0.2f * vv - uu);   // 0.25*0.02
            bool  sp = vn > 30.0f;
            float z  = sp ? 1.0f : 0.0f;
            v[j] = sp ? -65.0f : vn;
            u[j] = un + z * 6.0f;

            int gr2 = tileY * 16 + j + half * 8;
            int gc2 = tileX * 16 + n;
            out[base + (unsigned)(gr2 * IMG + gc2)] = (OutT)z;
        }
    };

    constexpr bool kAsync = HAS_ASYNC && (sizeof(InT) == 4);

    if constexpr (kAsync) {
        // ---- async double-buffered staging (fp32 input layer) ----
        // Zero both buffers once: border/pad slots are never touched by the
        // async engine afterwards, giving SAME zero padding for free.
        for (int idx = lane; idx < WSLICE; idx += 32) s0[idx] = 0.0f;
        asm volatile("s_wait_dscnt 0x0" ::: "memory");   // zeros landed before async writes

        // Per-lane element descriptors (t-invariant). Exactly NCHUNK async
        // instructions per stage: OOB lanes are redirected to the dump slot so
        // EXEC is always full and the ASYNCcnt bookkeeping is exact.
        int offv[NCHUNK]; int idxv[NCHUNK]; unsigned inbm = 0;
#pragma unroll
        for (int k = 0; k < NCHUNK; ++k) {
            int idx = lane + 32 * k;
            int r = idx / LDS_W, cc = idx - r * LDS_W;
            int gr = rowBase + r, gc = colBase + cc;
            bool inb = (idx < SLICE) && (cc < 18) &&
                       ((unsigned)gr < (unsigned)IMG) && ((unsigned)gc < (unsigned)IMG);
            offv[k] = inb ? (gr * IMG + gc) : 0;
            idxv[k] = idx;
            if (inb) inbm |= (1u << k);
        }

        auto issue = [&](int tt, float* nbuf) {
            const InT* p = in + (((size_t)(tt * BATCH + b)) << 16);
#pragma unroll
            for (int k = 0; k < NCHUNK; ++k) {
                bool ib = (inbm >> k) & 1u;
                const InT* g = p + offv[k];
                float* l = ib ? (nbuf + idxv[k]) : (dump + lane);
                __builtin_amdgcn_global_load_async_to_lds_b32(
                    (int*)(void*)g, (int*)(void*)l, 0, 0);
            }
        };

        issue(0, s0);
        for (int t = 0; t < T_STEPS; ++t) {
            const size_t base = ((size_t)(t * BATCH + b)) << 16;
            float* cur = (t & 1) ? s1 : s0;
            float* nxt = (t & 1) ? s0 : s1;
            if (t + 1 < T_STEPS) {
                issue(t + 1, nxt);
                // in-order completion: <=NCHUNK outstanding => stage t landed
                asm volatile("s_wait_asynccnt 0xc" ::: "memory");
            } else {
                asm volatile("s_wait_asynccnt 0x0" ::: "memory");
            }
            step(cur, base);
        }
    } else {
        // ---- scalar staging (u8 spike layers): wave-private slice, so a
        // wave-scope DS wait replaces block barriers entirely.
        for (int t = 0; t < T_STEPS; ++t) {
            const size_t base = ((size_t)(t * BATCH + b)) << 16;
            if (t + 1 < T_STEPS)
                __builtin_prefetch(in + base + ((size_t)BATCH << 16), 0, 1);
            for (int idx = lane; idx < SLICE; idx += 32) {
                int r  = idx / LDS_W;
                int cc = idx - r * LDS_W;
                int gr = rowBase + r, gc = colBase + cc;
                float val = 0.0f;
                if (cc < 18 && (unsigned)gr < (unsigned)IMG && (unsigned)gc < (unsigned)IMG)
                    val = (float)in[base + (unsigned)(gr * IMG + gc)];
                s0[idx] = val;
            }
            asm volatile("s_wait_dscnt 0x0" ::: "memory");  // stores visible to all lanes
            step(s0, base);
        }
    }
}

extern "C" void kernel_launch(void* const* d_in, const int* in_sizes, int n_in,
                              void* d_out, int out_size, void* d_ws, size_t ws_size,
                              hipStream_t stream) {
    const float* x = (const float*)d_in[0];   // [32,16,1,256,256] fp32
    const float* w = (const float*)d_in[1];   // [4,1,1,3,3] fp32
    float* outp = (float*)d_out;

    const size_t N = (size_t)T_STEPS * BATCH * IMG * IMG;   // 33,554,432
    unsigned char* s0 = (unsigned char*)d_ws;
    unsigned char* s1 = s0 + N;

    dim3 grid(512), block(256);
    izhi_layer<float,         unsigned char><<<grid, block, 0, stream>>>(x,  s0, w + 0);
    izhi_layer<unsigned char, unsigned char><<<grid, block, 0, stream>>>(s0, s1, w + 9);
    izhi_layer<unsigned char, unsigned char><<<grid, block, 0, stream>>>(s1, s0, w + 18);
    izhi_layer<unsigned char, float        ><<<grid, block, 0, stream>>>(s0, outp, w + 27);
}